// GNN_1881195675938
// MI455X (gfx1250) — compile-verified
//
#include <hip/hip_runtime.h>

#define IN_CH 128
#define HID   16
#define OUT_CH 64

typedef __attribute__((ext_vector_type(2))) float v2f;
typedef __attribute__((ext_vector_type(4))) float v4f;
typedef __attribute__((ext_vector_type(8))) float v8f;

// ---------------- degree / norm ----------------
__global__ void k_deg_init(float* deg, int n) {
    int i = blockIdx.x * blockDim.x + threadIdx.x;
    if (i < n) deg[i] = 1.0f;                 // self-loop contributes 1
}

__global__ void k_deg_count(const int* __restrict__ dst, float* deg, int E) {
    int e = blockIdx.x * blockDim.x + threadIdx.x;
    if (e < E) atomicAdd(deg + dst[e], 1.0f);
}

__global__ void k_dinv(float* deg, int n) {
    int i = blockIdx.x * blockDim.x + threadIdx.x;
    if (i < n) deg[i] = rsqrtf(deg[i]);       // deg >= 1 always
}

// ---------------- GEMM1: X[N,128] @ W1[128,16] -> H1[N,16] ----------------
// One wave per 16-row M tile. V_WMMA_F32_16X16X4_F32, 32 K-steps.
__global__ void __launch_bounds__(256) k_gemm1(const float* __restrict__ X,
                                               const float* __restrict__ W1,
                                               float* __restrict__ H1, int nTiles) {
    __shared__ float sWt[HID * 130];          // W1 transposed [n][k], padded stride
    for (int i = threadIdx.x; i < IN_CH * HID; i += 256) {
        int k = i >> 4, n = i & 15;
        sWt[n * 130 + k] = W1[i];
    }
    __syncthreads();

    int lane = threadIdx.x & 31;
    int wave = threadIdx.x >> 5;
    int tile = blockIdx.x * 8 + wave;
    if (tile >= nTiles) return;               // wave-uniform: EXEC stays full for WMMA

    int col  = lane & 15;                     // A row (M) / B col (N)
    int half = lane >> 4;                     // selects K 0,1 vs 2,3
    int kb   = half << 1;

    const float* ap = X + (tile * 16 + col) * IN_CH + kb;   // 8B aligned
    const float* bp = sWt + col * 130 + kb;                 // 8B aligned

    v8f acc = {};
    #pragma unroll
    for (int k = 0; k < IN_CH; k += 4) {
        v2f a = *(const v2f*)(ap + k);
        v2f b = *(const v2f*)(bp + k);
        acc = __builtin_amdgcn_wmma_f32_16x16x4_f32(false, a, false, b,
                                                    (short)0, acc, false, false);
    }

    // C/D layout: lane half -> rows 0..7 or 8..15, lane&15 -> column
    int rbase = tile * 16 + half * 8;
    float* op = H1 + rbase * HID + col;
    #pragma unroll
    for (int r = 0; r < 8; ++r) op[r * HID] = acc[r];
}

// ---------------- layer-1 self-loop + bias init ----------------
__global__ void k_init_agg1(const float* __restrict__ H1, const float* __restrict__ dinv,
                            const float* __restrict__ b1, float* __restrict__ agg1, int n) {
    int t = blockIdx.x * blockDim.x + threadIdx.x;
    if (t >= n * HID) return;
    int i = t >> 4, c = t & 15;
    float w = dinv[i]; w *= w;
    agg1[t] = H1[t] * w + b1[c];
}

// ---------------- layer-1 edge scatter: 4 threads/edge, 4 ch each ----------------
__global__ void k_scatter1(const int* __restrict__ ei, const float* __restrict__ dinv,
                           const float* __restrict__ H1, float* agg1, int E) {
    int t = blockIdx.x * blockDim.x + threadIdx.x;
    if (t >= E * 4) return;
    int e  = t >> 2;
    int c0 = (t & 3) << 2;
    int s = ei[e], d = ei[E + e];
    float w = dinv[s] * dinv[d];
    v4f m = *(const v4f*)(H1 + s * HID + c0);
    float* o = agg1 + d * HID + c0;
    atomicAdd(o + 0, m.x * w);
    atomicAdd(o + 1, m.y * w);
    atomicAdd(o + 2, m.z * w);
    atomicAdd(o + 3, m.w * w);
}

// ---------------- GEMM2: relu(agg1)[N,16] @ W2[16,64] -> H2[N,64] ----------------
// One wave per M tile; 4 N-tiles with A-fragment reuse; 4 WMMA steps each.
__global__ void __launch_bounds__(256) k_gemm2(const float* __restrict__ A,
                                               const float* __restrict__ W2,
                                               float* __restrict__ H2, int nTiles) {
    __shared__ float sWt[OUT_CH * 18];        // W2 transposed [n][k], padded
    for (int i = threadIdx.x; i < HID * OUT_CH; i += 256) {
        int k = i >> 6, n = i & 63;
        sWt[n * 18 + k] = W2[i];
    }
    __syncthreads();

    int lane = threadIdx.x & 31;
    int wave = threadIdx.x >> 5;
    int tile = blockIdx.x * 8 + wave;
    if (tile >= nTiles) return;

    int col  = lane & 15;
    int half = lane >> 4;
    int kb   = half << 1;

    const float* ap = A + (tile * 16 + col) * HID + kb;
    v2f afrag[4];
    #pragma unroll
    for (int s = 0; s < 4; ++s) {
        v2f a = *(const v2f*)(ap + s * 4);
        a.x = fmaxf(a.x, 0.0f);               // fused ReLU
        a.y = fmaxf(a.y, 0.0f);
        afrag[s] = a;
    }

    v8f acc[4] = {};
    #pragma unroll
    for (int t = 0; t < 4; ++t) {
        const float* bp = sWt + (t * 16 + col) * 18 + kb;
        #pragma unroll
        for (int s = 0; s < 4; ++s) {
            v2f b = *(const v2f*)(bp + s * 4);
            acc[t] = __builtin_amdgcn_wmma_f32_16x16x4_f32(false, afrag[s], false, b,
                                                           (short)0, acc[t], false, false);
        }
    }

    int rbase = tile * 16 + half * 8;
    #pragma unroll
    for (int t = 0; t < 4; ++t) {
        float* op = H2 + rbase * OUT_CH + t * 16 + col;
        #pragma unroll
        for (int r = 0; r < 8; ++r) op[r * OUT_CH] = acc[t][r];
    }
}

// ---------------- layer-2 self-loop + bias init (also un-poisons d_out) ----------------
__global__ void k_init_out(const float* __restrict__ H2, const float* __restrict__ dinv,
                           const float* __restrict__ b2, float* __restrict__ out, int n) {
    int t = blockIdx.x * blockDim.x + threadIdx.x;
    if (t >= n * OUT_CH) return;
    int i = t >> 6, c = t & 63;
    float w = dinv[i]; w *= w;
    out[t] = H2[t] * w + b2[c];
}

// ---------------- layer-2 edge scatter: 16 threads/edge, 4 ch each ----------------
__global__ void k_scatter2(const int* __restrict__ ei, const float* __restrict__ dinv,
                           const float* __restrict__ H2, float* out, int E) {
    long long t = (long long)blockIdx.x * blockDim.x + threadIdx.x;
    if (t >= (long long)E * 16) return;
    int e  = (int)(t >> 4);
    int c0 = ((int)t & 15) << 2;
    int s = ei[e], d = ei[E + e];
    float w = dinv[s] * dinv[d];
    v4f m = *(const v4f*)(H2 + s * OUT_CH + c0);
    float* o = out + d * OUT_CH + c0;
    atomicAdd(o + 0, m.x * w);
    atomicAdd(o + 1, m.y * w);
    atomicAdd(o + 2, m.z * w);
    atomicAdd(o + 3, m.w * w);
}

extern "C" void kernel_launch(void* const* d_in, const int* in_sizes, int n_in,
                              void* d_out, int out_size, void* d_ws, size_t ws_size,
                              hipStream_t stream) {
    const float* x  = (const float*)d_in[0];
    const int*   ei = (const int*)d_in[1];
    const float* W1 = (const float*)d_in[2];
    const float* b1 = (const float*)d_in[3];
    const float* W2 = (const float*)d_in[4];
    const float* b2 = (const float*)d_in[5];
    float* out = (float*)d_out;

    int N = in_sizes[0] / IN_CH;      // 100000
    int E = in_sizes[1] / 2;          // 3200000
    int nTiles = (N + 15) / 16;       // 6250 (N divisible by 16)

    float* ws   = (float*)d_ws;
    float* dinv = ws;                               // N
    float* H1   = dinv + N;                         // N*16
    float* agg1 = H1 + (size_t)N * HID;             // N*16
    float* H2   = agg1 + (size_t)N * HID;           // N*64

    const int B = 256;
    k_deg_init <<<(N + B - 1) / B, B, 0, stream>>>(dinv, N);
    k_deg_count<<<(E + B - 1) / B, B, 0, stream>>>(ei + E, dinv, E);
    k_dinv     <<<(N + B - 1) / B, B, 0, stream>>>(dinv, N);

    k_gemm1    <<<(nTiles + 7) / 8, B, 0, stream>>>(x, W1, H1, nTiles);
    k_init_agg1<<<((size_t)N * HID + B - 1) / B, B, 0, stream>>>(H1, dinv, b1, agg1, N);
    k_scatter1 <<<((size_t)E * 4 + B - 1) / B, B, 0, stream>>>(ei, dinv, H1, agg1, E);

    k_gemm2    <<<(nTiles + 7) / 8, B, 0, stream>>>(agg1, W2, H2, nTiles);
    k_init_out <<<((size_t)N * OUT_CH + B - 1) / B, B, 0, stream>>>(H2, dinv, b2, out, N);
    k_scatter2 <<<((size_t)E * 16 + B - 1) / B, B, 0, stream>>>(ei, dinv, H2, out, E);
}